// head_attention_84653805404708
// MI455X (gfx1250) — compile-verified
//
#include <hip/hip_runtime.h>

// Problem constants (reference: B,N,D,DH = 2,512,256,16)
#define BB 2
#define NN 512
#define DD 256
#define DHH 16

typedef __attribute__((ext_vector_type(2))) float v2f;
typedef __attribute__((ext_vector_type(8))) float v8f;

// ---------------------------------------------------------------------------
// Kernel 1: node projections  Q,K,V,Ni,Nj = x @ W* + b*   ([B*N,16] each)
// 81920 threads, one (proj,row,col) per thread. Tiny vs. the 512MB edge pass.
// ---------------------------------------------------------------------------
__global__ __launch_bounds__(256) void proj_kernel(
    const float* __restrict__ x,
    const float* __restrict__ Wq, const float* __restrict__ bq,
    const float* __restrict__ Wk, const float* __restrict__ bk,
    const float* __restrict__ Wv, const float* __restrict__ bv,
    const float* __restrict__ Wni, const float* __restrict__ bni,
    const float* __restrict__ Wnj, const float* __restrict__ bnj,
    float* __restrict__ ws)
{
    int t   = blockIdx.x * 256 + threadIdx.x;   // 0..81919
    int col = t & 15;
    int row = (t >> 4) & 1023;                  // b*N+i
    int p   = t >> 14;                          // 0..4 : Q,K,V,Ni,Nj
    const float* Wl[5] = {Wq, Wk, Wv, Wni, Wnj};
    const float* bl[5] = {bq, bk, bv, bni, bnj};
    const float* W  = Wl[p];
    const float* xr = x + (size_t)row * DD;
    float acc = bl[p][col];
#pragma unroll 4
    for (int d = 0; d < DD; ++d)
        acc = fmaf(xr[d], W[d * DHH + col], acc);
    ws[(size_t)p * (1024 * DHH) + row * DHH + col] = acc;
}

// ---------------------------------------------------------------------------
// Kernel 2 (dominant, bandwidth-bound): for each (b,i, 16-wide j-tile):
//   E      = e[b,i,j0:j0+16,:] @ We            (f32 WMMA 16x16x4, 64 k-steps)
//   e_new  = E + be + Ni[b,i,:] + Nj[b,j,:]    -> d_out   (NT stores)
//   S[b,i,j] = sum_h Q[b,i,h]*e_new*K[b,j,h] / 4  (shfl-reduce over h) -> ws
// One wave per tile; 4 waves/block; We staged in LDS in B-fragment layout.
// e is 512MB streamed once -> non-temporal loads so the 192MB L2 keeps the
// reused operands (We, Q/K/Ni/Nj, S) instead.
// ---------------------------------------------------------------------------
__global__ __launch_bounds__(128) void edge_kernel(
    const float* __restrict__ e,
    const float* __restrict__ We, const float* __restrict__ be,
    const float* __restrict__ Qp, const float* __restrict__ Kp,
    const float* __restrict__ Nip, const float* __restrict__ Njp,
    float* __restrict__ e_new, float* __restrict__ S)
{
    // B-fragment layout per ISA 7.12.2: VGPR0 lane L -> (K = 2*(L>>4)+k, N = L&15)
    //                                   VGPR1 lane L -> (K = 2*(L>>4)+k+1, N = L&15)
    __shared__ float lds_w[64 * 32 * 2];   // 16 KB: [iter][lane][2]
    for (int idx = threadIdx.x; idx < 64 * 32; idx += 128) {
        int it  = idx >> 5;
        int ln  = idx & 31;
        int col = ln & 15;
        int k   = it * 4 + ((ln >> 4) << 1);
        lds_w[idx * 2 + 0] = We[(k + 0) * DHH + col];
        lds_w[idx * 2 + 1] = We[(k + 1) * DHH + col];
    }
    __syncthreads();

    int lane = threadIdx.x & 31;
    int wave = threadIdx.x >> 5;
    int tile = blockIdx.x * 4 + wave;       // 0..32767
    int b    = tile >> 14;                  // 16384 tiles per batch
    int rem  = tile & 16383;
    int i    = rem >> 5;                    // 32 j-tiles per i
    int j0   = (rem & 31) << 4;

    int m    = lane & 15;                   // A-matrix row (local j) / C column h
    int half = lane >> 4;
    // A-fragment: lane m holds (K=k,k+1), lane m+16 holds (K=k+2,k+3) of row m
    const float* arow = e + (((size_t)b * NN + i) * NN + j0 + m) * DD + half * 2;
    const float* wp   = &lds_w[lane * 2];

    v8f c = {0.f, 0.f, 0.f, 0.f, 0.f, 0.f, 0.f, 0.f};
    for (int it0 = 0; it0 < 64; it0 += 16) {
        // straight-line streaming prefetch once per 16 k-steps
        __builtin_prefetch(arow + it0 * 4 + 256, 0, 0);
#pragma unroll
        for (int it1 = 0; it1 < 16; ++it1) {
            int it = it0 + it1;
            v2f a = __builtin_nontemporal_load((const v2f*)(arow + it * 4));
            v2f w = *(const v2f*)(wp + it * 64);    // ds_load_b64
            c = __builtin_amdgcn_wmma_f32_16x16x4_f32(false, a, false, w,
                                                      (short)0, c, false, false);
        }
    }

    // Epilogue: C VGPR r = rows (r, r+8), column h = lane&15
    int h  = m;
    int nb = b * NN;
    float qv   = Qp[(nb + i) * DHH + h];
    float base = be[h] + Nip[(nb + i) * DHH + h];
    float* eo  = e_new + (((size_t)nb + i) * NN + j0) * DHH;

#pragma unroll
    for (int r = 0; r < 8; ++r) {
        int   rl = r + half * 8;            // local row in tile
        int   j  = j0 + rl;
        float v  = c[r] + base + Njp[(nb + j) * DHH + h];
        __builtin_nontemporal_store(v, &eo[rl * DHH + h]);
        float p = v * qv * Kp[(nb + j) * DHH + h];
        p += __shfl_xor(p, 1);              // reduce over h within 16-lane half
        p += __shfl_xor(p, 2);
        p += __shfl_xor(p, 4);
        p += __shfl_xor(p, 8);
        if (h == 0)
            S[(size_t)(nb + i) * NN + j] = p * 0.25f;   // / sqrt(16)
    }
}

// ---------------------------------------------------------------------------
// Kernel 3a: softmax stats over axis i (per (b,j) column), online pass.
// Adjacent threads -> adjacent j -> coalesced column reads.
// ---------------------------------------------------------------------------
__global__ __launch_bounds__(256) void colsoftmax_kernel(
    const float* __restrict__ S, float* __restrict__ mbuf, float* __restrict__ sinv)
{
    int t = blockIdx.x * 256 + threadIdx.x; // 0..1023
    int b = t >> 9;
    int j = t & 511;
    const float* Sp = S + (size_t)b * NN * NN + j;
    float m = -3.402823e38f;
    float s = 0.f;
    for (int i = 0; i < NN; ++i) {
        float v  = Sp[(size_t)i * NN];
        float mn = fmaxf(m, v);
        s = s * __expf(m - mn) + __expf(v - mn);
        m = mn;
    }
    mbuf[t] = m;
    sinv[t] = 1.0f / s;
}

// ---------------------------------------------------------------------------
// Kernel 3b: out[b,i,h] = sum_j softmax_i(S)[b,i,j] * V[b,j,h]   (17 MFLOP)
// ---------------------------------------------------------------------------
__global__ __launch_bounds__(256) void av_kernel(
    const float* __restrict__ S, const float* __restrict__ mbuf,
    const float* __restrict__ sinv, const float* __restrict__ Vp,
    float* __restrict__ out)
{
    int t = blockIdx.x * 256 + threadIdx.x; // 0..16383
    int h = t & 15;
    int i = (t >> 4) & 511;
    int b = t >> 13;
    const float* Srow = S + ((size_t)b * NN + i) * NN;
    const float* mcol = mbuf + b * NN;
    const float* scol = sinv + b * NN;
    const float* Vb   = Vp + (size_t)b * NN * DHH + h;
    float acc = 0.f;
    for (int j = 0; j < NN; ++j) {
        float p = __expf(Srow[j] - mcol[j]) * scol[j];
        acc = fmaf(p, Vb[j * DHH], acc);
    }
    out[(b * NN + i) * DHH + h] = acc;
}

// ---------------------------------------------------------------------------
extern "C" void kernel_launch(void* const* d_in, const int* in_sizes, int n_in,
                              void* d_out, int out_size, void* d_ws, size_t ws_size,
                              hipStream_t stream)
{
    (void)in_sizes; (void)n_in; (void)out_size; (void)ws_size;

    const float* x   = (const float*)d_in[0];
    const float* e   = (const float*)d_in[1];
    const float* Wq  = (const float*)d_in[2];
    const float* bq  = (const float*)d_in[3];
    const float* Wk  = (const float*)d_in[4];
    const float* bk  = (const float*)d_in[5];
    const float* We  = (const float*)d_in[6];
    const float* be  = (const float*)d_in[7];
    const float* Wv  = (const float*)d_in[8];
    const float* bv  = (const float*)d_in[9];
    const float* Wni = (const float*)d_in[10];
    const float* bni = (const float*)d_in[11];
    const float* Wnj = (const float*)d_in[12];
    const float* bnj = (const float*)d_in[13];

    float* out   = (float*)d_out;                 // [2,512,16]
    float* e_new = out + BB * NN * DHH;           // [2,512,512,16]

    float* ws   = (float*)d_ws;
    float* Qp   = ws;                             // each [1024,16]
    float* Kp   = ws + 16384;
    float* Vp   = ws + 32768;
    float* Nip  = ws + 49152;
    float* Njp  = ws + 65536;
    float* S    = ws + 81920;                     // [2,512,512] raw scores
    float* mbuf = ws + 81920 + 524288;            // [1024]
    float* sinv = mbuf + 1024;                    // [1024]

    proj_kernel<<<320, 256, 0, stream>>>(x, Wq, bq, Wk, bk, Wv, bv,
                                         Wni, bni, Wnj, bnj, ws);
    edge_kernel<<<8192, 128, 0, stream>>>(e, We, be, Qp, Kp, Nip, Njp, e_new, S);
    colsoftmax_kernel<<<4, 256, 0, stream>>>(S, mbuf, sinv);
    av_kernel<<<64, 256, 0, stream>>>(S, mbuf, sinv, Vp, out);
}